// knrm_8727373545844
// MI455X (gfx1250) — compile-verified
//
#include <hip/hip_runtime.h>
#include <hip/hip_bf16.h>

typedef float v2f __attribute__((ext_vector_type(2)));
typedef float v8f __attribute__((ext_vector_type(8)));

#define B_    128
#define LQ_   20
#define LD_   512
#define D_    300
#define NB_   21
#define MPAD  32      // M padded to 2 WMMA tiles

// ---------------------------------------------------------------------------
// ds_swizzle_b32 XOR butterfly (group-of-32 mode: and=0x1f, or=0, xor=MASK):
// one DS op per exchange, no bounds-check cndmask like __shfl_xor.
// ---------------------------------------------------------------------------
template <int MASK>
__device__ __forceinline__ float swz_xor(float v) {
  return __int_as_float(
      __builtin_amdgcn_ds_swizzle(__float_as_int(v), 0x1f | (MASK << 10)));
}

__device__ __forceinline__ float wave_sum32(float v) {
  v += swz_xor<16>(v);
  v += swz_xor<8>(v);
  v += swz_xor<4>(v);
  v += swz_xor<2>(v);
  v += swz_xor<1>(v);
  return v;
}

// ---------------------------------------------------------------------------
// Gather + padding-idx-0 + L2-normalize query embeddings -> qn[b][m][k]
// (row-major, rows 20..31 remain zero from the memset)
// ---------------------------------------------------------------------------
__global__ void __launch_bounds__(128) knrm_norm_q(const int* __restrict__ tok,
                                                   const float* __restrict__ emb,
                                                   float* __restrict__ qn) {
  const int t = blockIdx.x;               // 0 .. B*LQ-1
  const int b = t / LQ_, m = t % LQ_;
  const int tid = threadIdx.x;
  const int idx = tok[t];
  const float z = (idx == 0) ? 0.0f : 1.0f;   // padding_idx = 0 -> zero vector
  const float* e = emb + (size_t)idx * D_;

  float v0 = 0.f, v1 = 0.f, v2 = 0.f;
  if (tid        < D_) v0 = e[tid]       * z;
  if (tid + 128  < D_) v1 = e[tid + 128] * z;
  if (tid + 256  < D_) v2 = e[tid + 256] * z;

  float ss = wave_sum32(v0*v0 + v1*v1 + v2*v2);
  __shared__ float red[4];
  if ((tid & 31) == 0) red[tid >> 5] = ss;
  __syncthreads();
  ss = red[0] + red[1] + red[2] + red[3];
  const float inv = 1.0f / fmaxf(sqrtf(ss), 1e-12f);

  float* o = qn + ((size_t)b * MPAD + m) * D_;
  if (tid        < D_) o[tid]       = v0 * inv;
  if (tid + 128  < D_) o[tid + 128] = v1 * inv;
  if (tid + 256  < D_) o[tid + 256] = v2 * inv;
}

// ---------------------------------------------------------------------------
// Gather + normalize doc embeddings; store K-pair-interleaved, K-major:
//   element (k, n) of batch b  ->  dT[b*D*LD + ((k>>1)*LD + n)*2 + (k&1)]
// so the WMMA B-operand is ONE aligned b64 load per step and lanes 0..15
// cover a contiguous 128B line.
// ---------------------------------------------------------------------------
__global__ void __launch_bounds__(128) knrm_norm_d(const int* __restrict__ tok,
                                                   const float* __restrict__ emb,
                                                   float* __restrict__ dT) {
  const int t = blockIdx.x;               // 0 .. B*LD-1
  const int b = t / LD_, n = t % LD_;
  const int tid = threadIdx.x;
  const int idx = tok[t];
  const float z = (idx == 0) ? 0.0f : 1.0f;
  const float* e = emb + (size_t)idx * D_;

  float v0 = 0.f, v1 = 0.f, v2 = 0.f;
  if (tid        < D_) v0 = e[tid]       * z;
  if (tid + 128  < D_) v1 = e[tid + 128] * z;
  if (tid + 256  < D_) v2 = e[tid + 256] * z;

  float ss = wave_sum32(v0*v0 + v1*v1 + v2*v2);
  __shared__ float red[4];
  if ((tid & 31) == 0) red[tid >> 5] = ss;
  __syncthreads();
  ss = red[0] + red[1] + red[2] + red[3];
  const float inv = 1.0f / fmaxf(sqrtf(ss), 1e-12f);

  float* o = dT + (size_t)b * D_ * LD_;
  #pragma unroll
  for (int s = 0; s < 3; ++s) {
    const int k = tid + s * 128;
    const float v = (s == 0) ? v0 : (s == 1) ? v1 : v2;
    if (k < D_) o[((size_t)(k >> 1) * LD_ + n) * 2 + (k & 1)] = v * inv;
  }
}

// ---------------------------------------------------------------------------
// Fused cosine-sim GEMM (V_WMMA_F32_16X16X4_F32) + RBF binning + pooling-sum.
// Block = 128 threads = 4 waves. Grid = B * 2(mtile) * 8(ntile group).
// A-operand: lane(0..15)=M, VGPR pair = {K=k0, K=k0+1}, upper half at K+2
// (from LDS). B-operand: lane(0..15)=N, same K striping (one b64 from dT).
// ---------------------------------------------------------------------------
__global__ void __launch_bounds__(128)
knrm_sim_bin(const float* __restrict__ qn, const float* __restrict__ dT,
             const float* __restrict__ mask_d, const float* __restrict__ mu,
             const float* __restrict__ sigma, float* __restrict__ pool) {
  __shared__ float lds_a[16 * D_];
  __shared__ float lds_mu[NB_];
  __shared__ float lds_c[NB_];          // -log2(e) / (2*sigma^2)

  const int bid   = blockIdx.x;
  const int b     = bid >> 4;
  const int rem   = bid & 15;
  const int mtile = rem >> 3;      // 0..1
  const int ng    = rem & 7;       // 0..7 (group of 4 n-tiles)
  const int tid   = threadIdx.x;
  const int w     = tid >> 5;
  const int lane  = tid & 31;
  const int l15   = lane & 15;
  const int half  = lane >> 4;

  // Stage the 16x300 A tile (contiguous in qn) into LDS.
  const float* aSrc = qn + ((size_t)b * MPAD + mtile * 16) * D_;
  for (int i = tid; i < 16 * D_; i += 128) lds_a[i] = aSrc[i];
  if (tid < NB_) {
    lds_mu[tid] = mu[tid];
    const float s = sigma[tid];
    lds_c[tid] = -1.4426950408889634f / (2.0f * s * s);
  }
  __syncthreads();

  const int n  = (ng * 4 + w) * 16 + l15;
  const int kh = half << 1;                            // K offset of upper half
  // B base: pair-row (k>>1)=half-adjusted, column n, pair-interleaved layout.
  const float* bp = dT + (size_t)b * D_ * LD_ + ((size_t)half * LD_ + n) * 2;

  v8f c = {};
  #pragma unroll 5
  for (int k = 0; k < D_; k += 4) {
    const int k0 = k + kh;
    v2f a;
    a.x = lds_a[l15 * D_ + k0];
    a.y = lds_a[l15 * D_ + k0 + 1];
    const v2f bb = *(const v2f*)(bp + (size_t)(k >> 1) * (2 * LD_));
    c = __builtin_amdgcn_wmma_f32_16x16x4_f32(false, a, false, bb,
                                              (short)0, c, false, false);
  }

  // RBF soft binning; butterfly-reduce the 16 doc columns per half-wave with
  // single-op ds_swizzle XOR exchanges; accumulate into pool[b][m][bin].
  const float md = mask_d[b * LD_ + n];
  #pragma unroll
  for (int r = 0; r < 8; ++r) {
    const int m   = mtile * 16 + r + (half << 3);     // C layout: VGPR r -> M=r / M=r+8
    const float s = c[r];
    const bool valid = (m < LQ_);
    for (int j = 0; j < NB_; ++j) {
      const float dm = s - lds_mu[j];
      float v = __builtin_amdgcn_exp2f(dm * dm * lds_c[j]) * md;
      v += swz_xor<1>(v);
      v += swz_xor<2>(v);
      v += swz_xor<4>(v);
      v += swz_xor<8>(v);
      if (l15 == 0 && valid)
        atomicAdd(&pool[((size_t)b * LQ_ + m) * NB_ + j], v);
    }
  }
}

// ---------------------------------------------------------------------------
// log-pool over LQ, dot with dense_w, tanh. One wave per batch, lane = bin.
// ---------------------------------------------------------------------------
__global__ void __launch_bounds__(32)
knrm_final(const float* __restrict__ pool, const float* __restrict__ mask_q,
           const float* __restrict__ dw, const float* __restrict__ db,
           float* __restrict__ out) {
  const int b = blockIdx.x;
  const int j = threadIdx.x;
  float f = 0.0f;
  if (j < NB_) {
    for (int m = 0; m < LQ_; ++m) {
      const float p = pool[((size_t)b * LQ_ + m) * NB_ + j];
      // ln(x) = log2(x) * ln(2)
      f += __builtin_amdgcn_logf(fmaxf(p, 1e-10f)) * 0.6931471805599453f
           * mask_q[b * LQ_ + m] * 0.01f;
    }
    f *= dw[j];
  }
  f = wave_sum32(f);
  if (j == 0) out[b] = tanhf(f + db[0]);
}

// ---------------------------------------------------------------------------
extern "C" void kernel_launch(void* const* d_in, const int* in_sizes, int n_in,
                              void* d_out, int out_size, void* d_ws, size_t ws_size,
                              hipStream_t stream) {
  const int*   inputs_q = (const int*)  d_in[0];
  const int*   inputs_d = (const int*)  d_in[1];
  const float* mask_q   = (const float*)d_in[2];
  const float* mask_d   = (const float*)d_in[3];
  const float* emb      = (const float*)d_in[4];
  const float* mu       = (const float*)d_in[5];
  const float* sigma    = (const float*)d_in[6];
  const float* dense_w  = (const float*)d_in[7];
  const float* dense_b  = (const float*)d_in[8];
  float*       out      = (float*)d_out;

  // Workspace layout (bytes)
  const size_t qn_bytes   = (size_t)B_ * MPAD * D_ * sizeof(float);   // 4.9 MB
  const size_t dT_bytes   = (size_t)B_ * D_ * LD_ * sizeof(float);    // 78.6 MB
  const size_t pool_bytes = (size_t)B_ * LQ_ * NB_ * sizeof(float);   // 0.2 MB
  char* ws = (char*)d_ws;
  float* qn   = (float*)(ws);
  float* dT   = (float*)(ws + qn_bytes);
  float* pool = (float*)(ws + qn_bytes + dT_bytes);
  (void)ws_size; (void)in_sizes; (void)n_in; (void)out_size;

  // Idempotent per-launch init (graph-capture safe).
  hipMemsetAsync(qn,   0, qn_bytes,   stream);   // zero padding rows 20..31
  hipMemsetAsync(pool, 0, pool_bytes, stream);   // atomic accumulators

  knrm_norm_q <<<B_ * LQ_, 128, 0, stream>>>(inputs_q, emb, qn);
  knrm_norm_d <<<B_ * LD_, 128, 0, stream>>>(inputs_d, emb, dT);
  knrm_sim_bin<<<B_ * 16,  128, 0, stream>>>(qn, dT, mask_d, mu, sigma, pool);
  knrm_final  <<<B_,        32, 0, stream>>>(pool, mask_q, dense_w, dense_b, out);
}